// GCN_87892210746076
// MI455X (gfx1250) — compile-verified
//
#include <hip/hip_runtime.h>

typedef __attribute__((ext_vector_type(2))) float v2f;
typedef __attribute__((ext_vector_type(8))) float v8f;

#define N_NODES   100000
#define N_EDGES   1600000
#define IN_FEAT   128
#define OUT_FEAT  64
#define RRELU_SLOPE 0.2291666666666667f   // 11/48

// ---------------------------------------------------------------------------
// GEMM: C[M x 64] = act(A[M x K]) * W[K x 64], act = identity or rrelu.
// One wave per 16-row tile; 4 x v8f accumulators cover all 64 columns.
// V_WMMA_F32_16X16X4_F32 operand layouts (CDNA5 ISA 7.12.2):
//   A 16x4:  lane l/half h holds {A[l][k0+2h], A[l][k0+2h+1]}  -> one b64 load
//   B 4x16:  lane l/half h holds {W[k0+2h][n], W[k0+2h+1][n]}  -> staged in LDS
//            pre-interleaved as float2 so it is one aligned ds_load_b64
//   C 16x16: vgpr j = row (j | j+8h) at col = l
// ---------------------------------------------------------------------------
template <int K, bool RRELU_IN>
__global__ __launch_bounds__(256) void gemm64_wmma(const float* __restrict__ A,
                                                   const float* __restrict__ W,
                                                   float* __restrict__ C, int M) {
  // lw[p*64 + n] = { W[2p][n], W[2p+1][n] }   (p = K-pair index)
  __shared__ v2f lw[(K / 2) * OUT_FEAT];
  const int tid = threadIdx.x;
  #pragma unroll 4
  for (int i = tid; i < (K / 2) * OUT_FEAT; i += 256) {
    const int p = i >> 6;        // pair row
    const int n = i & 63;        // column
    v2f w;
    w[0] = W[(2 * p) * OUT_FEAT + n];
    w[1] = W[(2 * p + 1) * OUT_FEAT + n];
    lw[i] = w;
  }
  __syncthreads();

  const int wave = tid >> 5;
  const int lane = tid & 31;
  const int l    = lane & 15;
  const int half = lane >> 4;
  const int mtile = blockIdx.x * 8 + wave;
  if (mtile * 16 >= M) return;               // wave-uniform exit (after barrier)

  int row = mtile * 16 + l;
  if (row >= M) row = M - 1;                 // clamp loads; stores predicated below
  const float* __restrict__ arow = A + (size_t)row * K;

  v8f acc0 = {}, acc1 = {}, acc2 = {}, acc3 = {};

  for (int k0 = 0; k0 < K; k0 += 4) {
    v2f a = *(const v2f*)(arow + k0 + 2 * half);   // contiguous -> global_load_b64
    if (RRELU_IN) {
      a[0] = (a[0] >= 0.0f) ? a[0] : a[0] * RRELU_SLOPE;
      a[1] = (a[1] >= 0.0f) ? a[1] : a[1] * RRELU_SLOPE;
    }
    const v2f* __restrict__ bp = &lw[((k0 >> 1) + half) * OUT_FEAT + l];
    acc0 = __builtin_amdgcn_wmma_f32_16x16x4_f32(false, a, false, bp[0],  (short)0, acc0, false, false);
    acc1 = __builtin_amdgcn_wmma_f32_16x16x4_f32(false, a, false, bp[16], (short)0, acc1, false, false);
    acc2 = __builtin_amdgcn_wmma_f32_16x16x4_f32(false, a, false, bp[32], (short)0, acc2, false, false);
    acc3 = __builtin_amdgcn_wmma_f32_16x16x4_f32(false, a, false, bp[48], (short)0, acc3, false, false);
  }

  const int m0 = mtile * 16 + 8 * half;
  #pragma unroll
  for (int j = 0; j < 8; ++j) {
    const int r = m0 + j;
    if (r < M) {
      float* crow = C + (size_t)r * OUT_FEAT + l;
      crow[0]  = acc0[j];
      crow[16] = acc1[j];
      crow[32] = acc2[j];
      crow[48] = acc3[j];
    }
  }
}

// ---------------------------------------------------------------------------
// Edge scatter-sum: out[dst[e], :] += H[src[e], :].  16 threads per edge,
// float4 gather + 4 global_atomic_add_f32 (dest arrays are L2-resident).
// ---------------------------------------------------------------------------
__global__ __launch_bounds__(256) void scatter_add(const float* __restrict__ H,
                                                   const int* __restrict__ src,
                                                   const int* __restrict__ dst,
                                                   float* __restrict__ out) {
  const unsigned t = blockIdx.x * 256u + threadIdx.x;
  const unsigned e = t >> 4;
  if (e >= (unsigned)N_EDGES) return;
  const unsigned c = (t & 15u) << 2;
  const int s = src[e];
  const int d = dst[e];
  const float4 v = *(const float4*)(H + (size_t)s * OUT_FEAT + c);
  float* o = out + (size_t)d * OUT_FEAT + c;
  atomicAdd(o + 0, v.x);
  atomicAdd(o + 1, v.y);
  atomicAdd(o + 2, v.z);
  atomicAdd(o + 3, v.w);
}

__global__ __launch_bounds__(256) void zero_f4(float4* __restrict__ p, unsigned n4) {
  const unsigned i = blockIdx.x * 256u + threadIdx.x;
  if (i < n4) p[i] = make_float4(0.f, 0.f, 0.f, 0.f);
}

__global__ __launch_bounds__(256) void rrelu_inplace_f4(float4* __restrict__ p, unsigned n4) {
  const unsigned i = blockIdx.x * 256u + threadIdx.x;
  if (i < n4) {
    float4 v = p[i];
    v.x = (v.x >= 0.f) ? v.x : v.x * RRELU_SLOPE;
    v.y = (v.y >= 0.f) ? v.y : v.y * RRELU_SLOPE;
    v.z = (v.z >= 0.f) ? v.z : v.z * RRELU_SLOPE;
    v.w = (v.w >= 0.f) ? v.w : v.w * RRELU_SLOPE;
    p[i] = v;
  }
}

// ---------------------------------------------------------------------------
// inputs (setup_inputs order): feat f32[100000*128], src i[1.6M], dst i[1.6M],
//                              w1 f32[128*64], w2 f32[64*64]
// output: f32[100000*64]
// ---------------------------------------------------------------------------
extern "C" void kernel_launch(void* const* d_in, const int* in_sizes, int n_in,
                              void* d_out, int out_size, void* d_ws, size_t ws_size,
                              hipStream_t stream) {
  const float* feat = (const float*)d_in[0];
  const int*   src  = (const int*)d_in[1];
  const int*   dst  = (const int*)d_in[2];
  const float* w1   = (const float*)d_in[3];
  const float* w2   = (const float*)d_in[4];
  float* out = (float*)d_out;

  float* H1 = (float*)d_ws;                        // 25.6 MB
  float* H2 = H1 + (size_t)N_NODES * OUT_FEAT;     // 25.6 MB

  const unsigned HN   = (unsigned)N_NODES * OUT_FEAT;  // 6,400,000 floats
  const unsigned HN4  = HN / 4;
  const int mtiles     = (N_NODES + 15) / 16;          // 6250
  const int gemmBlocks = (mtiles + 7) / 8;             // 782
  const unsigned zeroBlocks    = (HN4 + 255) / 256;
  const unsigned scatterBlocks = ((unsigned)N_EDGES * 16u + 255u) / 256u;
  dim3 blk(256);

  // h1 = feat @ w1
  gemm64_wmma<IN_FEAT, false><<<gemmBlocks, blk, 0, stream>>>(feat, w1, H1, N_NODES);
  // h2 = segment_sum(h1[src], dst)
  zero_f4<<<zeroBlocks, blk, 0, stream>>>((float4*)H2, HN4);
  scatter_add<<<scatterBlocks, blk, 0, stream>>>(H1, src, dst, H2);
  // h3 = rrelu(h2) @ w2   (rrelu fused into A-load)
  gemm64_wmma<OUT_FEAT, true><<<gemmBlocks, blk, 0, stream>>>(H2, w2, H1, N_NODES);
  // out = rrelu(segment_sum(h3[src], dst))
  zero_f4<<<zeroBlocks, blk, 0, stream>>>((float4*)out, HN4);
  scatter_add<<<scatterBlocks, blk, 0, stream>>>(H1, src, dst, out);
  rrelu_inplace_f4<<<zeroBlocks, blk, 0, stream>>>((float4*)out, HN4);
}